// Encoder_6313601925376
// MI455X (gfx1250) — compile-verified
//
#include <hip/hip_runtime.h>

typedef __attribute__((ext_vector_type(2))) float v2f;
typedef __attribute__((ext_vector_type(8))) float v8f;

#define HDIM 128
#define KDIM 256

// ---------------- degree / normalization ----------------
__global__ __launch_bounds__(256) void k_deg_init(float* deg, int N) {
    int i = blockIdx.x * 256 + threadIdx.x;
    if (i < N) deg[i] = 1.0f;  // self-loop contributes 1 to every destination
}

__global__ __launch_bounds__(256) void k_deg_count(const long long* __restrict__ col,
                                                   float* deg, int E) {
    int i = blockIdx.x * 256 + threadIdx.x;
    if (i < E) atomicAdd(&deg[(int)col[i]], 1.0f);
}

__global__ __launch_bounds__(256) void k_rsqrt_inplace(float* deg, int N) {
    int i = blockIdx.x * 256 + threadIdx.x;
    if (i < N) deg[i] = rsqrtf(deg[i]);  // deg buffer becomes dinv
}

// ---------------- h = x @ W via V_WMMA_F32_16X16X4_F32 ----------------
// Block: 256 threads = 8 waves. Block owns 16 rows of x (staged in LDS,
// stride 257 floats for bank-conflict-free A reads). Wave w owns output
// columns [16w, 16w+16).
__global__ __launch_bounds__(256) void k_gemm(const float* __restrict__ x,
                                              const float* __restrict__ W,
                                              float* __restrict__ h) {
    __shared__ float lds[16 * 257];
    const int m0 = blockIdx.x * 16;
    const int t  = threadIdx.x;

    // cooperative, coalesced stage of the 16x256 x-strip
    #pragma unroll
    for (int i = t; i < 16 * 256; i += 256) {
        int r = i >> 8, c = i & 255;
        lds[r * 257 + c] = x[(size_t)(m0 + r) * KDIM + c];
    }
    __syncthreads();

    const int wave = t >> 5;
    const int lane = t & 31;
    const int half = lane >> 4;   // selects K pair {0,1} vs {2,3}
    const int lr   = lane & 15;   // row (A) / col (B) within tile
    const int nb   = wave * 16;

    v8f acc = {};
    const float* wp    = W + nb + lr;          // B fragment base: W[k][nb+lr]
    const float* ap    = lds + lr * 257;       // A fragment base: x[m0+lr][k]

    #pragma unroll 4
    for (int k = 0; k < KDIM; k += 4) {
        v2f a, bf;
        a.x  = ap[k + 2 * half];
        a.y  = ap[k + 2 * half + 1];
        bf.x = wp[(size_t)(k + 2 * half) * HDIM];
        bf.y = wp[(size_t)(k + 2 * half + 1) * HDIM];
        // (neg_a, A, neg_b, B, c_mod, C, reuse_a, reuse_b)
        acc = __builtin_amdgcn_wmma_f32_16x16x4_f32(
            false, a, false, bf, (short)0, acc, false, false);
    }

    // D layout: VGPR v -> row m0 + v + 8*half, col nb + lr
    float* hp = h + (size_t)(m0 + 8 * half) * HDIM + nb + lr;
    #pragma unroll
    for (int v = 0; v < 8; ++v) hp[(size_t)v * HDIM] = acc[v];
}

// ---------------- out = b + h * dinv^2  (bias + self-loop message) ----------------
__global__ __launch_bounds__(256) void k_init_out(const float* __restrict__ h,
                                                  const float* __restrict__ dinv,
                                                  const float* __restrict__ b,
                                                  float* __restrict__ out, int total) {
    int idx = blockIdx.x * 256 + threadIdx.x;
    if (idx < total) {
        int node = idx >> 7;          // HDIM == 128
        int c    = idx & 127;
        float dv = dinv[node];
        out[idx] = b[c] + h[idx] * dv * dv;
    }
}

// ---------------- edge scatter: one wave per edge, float4 per lane ----------------
__global__ __launch_bounds__(256) void k_scatter(const long long* __restrict__ ei,
                                                 const float* __restrict__ h,
                                                 const float* __restrict__ dinv,
                                                 float* out, int E) {
    int gid  = blockIdx.x * blockDim.x + threadIdx.x;
    int wv   = gid >> 5;
    int lane = gid & 31;
    int nw   = (gridDim.x * blockDim.x) >> 5;
    for (int e = wv; e < E; e += nw) {
        int s = (int)ei[e];          // source (row)
        int d = (int)ei[E + e];      // destination (col)
        float norm = dinv[s] * dinv[d];
        float4 v = ((const float4*)(h + (size_t)s * HDIM))[lane];  // 512B coalesced gather
        float* od = out + (size_t)d * HDIM + lane * 4;
        atomicAdd(od + 0, v.x * norm);
        atomicAdd(od + 1, v.y * norm);
        atomicAdd(od + 2, v.z * norm);
        atomicAdd(od + 3, v.w * norm);
    }
}

// ---------------- PReLU (per-channel alpha), in place ----------------
__global__ __launch_bounds__(256) void k_prelu(float* out, const float* __restrict__ alpha,
                                               int total) {
    int idx = blockIdx.x * 256 + threadIdx.x;
    if (idx < total) {
        float v = out[idx];
        float a = alpha[idx & 127];
        out[idx] = (v >= 0.0f) ? v : a * v;
    }
}

extern "C" void kernel_launch(void* const* d_in, const int* in_sizes, int n_in,
                              void* d_out, int out_size, void* d_ws, size_t ws_size,
                              hipStream_t stream) {
    (void)n_in; (void)out_size; (void)ws_size;
    const float*     x     = (const float*)d_in[0];
    const long long* ei    = (const long long*)d_in[1];   // [2, E] row-major int64
    const float*     W     = (const float*)d_in[2];
    const float*     b     = (const float*)d_in[3];
    const float*     alpha = (const float*)d_in[4];
    float*           out   = (float*)d_out;

    const int N = in_sizes[0] / KDIM;   // 100000
    const int E = in_sizes[1] / 2;      // 1600000

    char*  ws  = (char*)d_ws;
    float* deg = (float*)ws;                                 // N floats (becomes dinv)
    size_t off = ((size_t)N * 4 + 255) & ~(size_t)255;
    float* h   = (float*)(ws + off);                         // N*HDIM floats

    const int total = N * HDIM;

    k_deg_init     <<<(N + 255) / 256,     256, 0, stream>>>(deg, N);
    k_deg_count    <<<(E + 255) / 256,     256, 0, stream>>>(ei + (size_t)E, deg, E);
    k_rsqrt_inplace<<<(N + 255) / 256,     256, 0, stream>>>(deg, N);
    k_gemm         <<<N / 16,              256, 0, stream>>>(x, W, h);
    k_init_out     <<<(total + 255) / 256, 256, 0, stream>>>(h, deg, b, out, total);
    k_scatter      <<<2048,                256, 0, stream>>>(ei, h, deg, out, E);
    k_prelu        <<<(total + 255) / 256, 256, 0, stream>>>(out, alpha, total);
}